// LorentzBatchNorm_47278999995076
// MI455X (gfx1250) — compile-verified
//
#include <hip/hip_runtime.h>
#include <math.h>

// Problem constants (from reference): x[64, 64*64, 128] f32
#define BSZ    64
#define NPTS   4096          // H*W
#define CH     128
#define EPS_F   1e-5f
#define CLAMP_F 1e-8f

typedef float v2f __attribute__((ext_vector_type(2)));
typedef float v8f __attribute__((ext_vector_type(8)));

// ---- pass configs ----
#define WPB_B   32                    // waves per batch, mean pass
#define PTS_W_B (NPTS / WPB_B)        // 128 points per wave (4 per WMMA step)
#define WPB_D   512                   // waves per batch, var/output passes
#define PTS_W_D (NPTS / WPB_D)        // 8 points per wave

// ---- workspace layout (float offsets) ----
#define OFF_WSUM  0u
#define SZ_WSUM   (BSZ * WPB_B * CH)          // 262144 floats
#define OFF_MEAN  (OFF_WSUM + SZ_WSUM)
#define SZ_MEAN   (BSZ * CH)                  // 8192
#define OFF_VPART (OFF_MEAN + SZ_MEAN)
#define SZ_VPART  (BSZ * WPB_D)               // 32768
#define OFF_G     (OFF_VPART + SZ_VPART)      // + BSZ floats
// total ~ 303 K floats ~= 1.2 MB

__device__ __forceinline__ float wave_sum(float v) {
  v += __shfl_xor(v, 1, 32);
  v += __shfl_xor(v, 2, 32);
  v += __shfl_xor(v, 4, 32);
  v += __shfl_xor(v, 8, 32);
  v += __shfl_xor(v, 16, 32);
  return v;
}

// =====================================================================
// Pass 1: per-wave partial channel sums via V_WMMA_F32_16X16X4_F32.
// D = ones[16x4] * B[4x16] + C : every row of D is the 4-point column sum,
// so each WMMA accumulates 4 points x 16 channels. With A == all-ones the
// result is invariant to the K->lane assignment of B; we only rely on
// column n == lane%16 (documented B/C/D striping).
// =====================================================================
__global__ void k_mean_partial(const float* __restrict__ x, float* __restrict__ ws) {
  const int wave = (blockIdx.x * blockDim.x + threadIdx.x) >> 5;   // 0 .. BSZ*WPB_B-1
  const int lane = threadIdx.x & 31;
  const int b  = wave / WPB_B;
  const int wb = wave % WPB_B;
  const float* __restrict__ xb = x + (size_t)b * NPTS * CH;
  const int p0 = wb * PTS_W_B;
  const int n  = lane & 15;          // column (channel within 16-group)
  const int kb = (lane >> 4) * 2;    // k rows held by this half-wave

  v2f a; a[0] = 1.0f; a[1] = 1.0f;   // all-ones A matrix (16x4)
  v8f acc[8];
  #pragma unroll
  for (int g = 0; g < 8; ++g) {
    v8f z = {0.f,0.f,0.f,0.f,0.f,0.f,0.f,0.f};
    acc[g] = z;
  }

  for (int it = 0; it < PTS_W_B; it += 4) {
    const float* __restrict__ xp = xb + (size_t)(p0 + it) * CH;
    __builtin_prefetch(xp + 4 * CH + lane * 4, 0, 0);   // next 4-point tile
    #pragma unroll
    for (int g = 0; g < 8; ++g) {
      v2f bm;
      bm[0] = xp[(kb + 0) * CH + g * 16 + n];
      bm[1] = xp[(kb + 1) * CH + g * 16 + n];
      acc[g] = __builtin_amdgcn_wmma_f32_16x16x4_f32(
          /*neg_a=*/false, a, /*neg_b=*/false, bm,
          /*c_mod=*/(short)0, acc[g], /*reuse_a=*/false, /*reuse_b=*/false);
    }
  }

  float* __restrict__ out = ws + OFF_WSUM + ((size_t)b * WPB_B + wb) * CH;
  if (lane < 16) {
    #pragma unroll
    for (int g = 0; g < 8; ++g) out[g * 16 + lane] = acc[g][0];  // row M=0 of D
  }
}

// =====================================================================
// Finalize mean: sum WPB_B partials, /N, project onto hyperboloid.
// One block per batch, 128 threads (one per channel). Deterministic.
// =====================================================================
__global__ void k_mean_finalize(float* __restrict__ ws) {
  __shared__ float sm[CH];
  __shared__ float sscale;
  const int b = blockIdx.x, t = threadIdx.x;
  const float* __restrict__ wsum = ws + OFF_WSUM + (size_t)b * WPB_B * CH;
  float s = 0.f;
  #pragma unroll 4
  for (int w = 0; w < WPB_B; ++w) s += wsum[w * CH + t];
  const float m = s * (1.0f / (float)NPTS);
  sm[t] = m * m;
  __syncthreads();
  #pragma unroll
  for (int off = 64; off > 0; off >>= 1) {
    if (t < off) sm[t] += sm[t + off];
    __syncthreads();
  }
  if (t == 0) {
    // -linner(m,m) = 2*m0^2 - sum(m^2);  t==0 holds m0
    const float q = 2.f * m * m - sm[0];
    sscale = 1.0f / sqrtf(fmaxf(q, CLAMP_F));
  }
  __syncthreads();
  ws[OFF_MEAN + (size_t)b * CH + t] = m * sscale;
}

// =====================================================================
// Shared per-point math: logmap at mean + parallel transport to origin.
// Wave-per-point: lane holds channels [4*lane, 4*lane+4) as float4.
// =====================================================================
__device__ __forceinline__ float4 tangent_at_origin(const float* __restrict__ xp,
                                                    float4 mv, float mean0, int lane) {
  float4 xv = *(const float4*)(xp + 4 * lane);
  const float x0 = __shfl(xv.x, 0, 32);                     // channel 0
  float pd = xv.x * mv.x + xv.y * mv.y + xv.z * mv.z + xv.w * mv.w;
  const float lin = wave_sum(pd) - 2.f * mean0 * x0;        // linner(mean,x)
  const float alpha = fmaxf(-lin, 1.0f + 1e-7f);
  float4 u;
  u.x = xv.x - alpha * mv.x;  u.y = xv.y - alpha * mv.y;
  u.z = xv.z - alpha * mv.z;  u.w = xv.w - alpha * mv.w;
  const float u0 = x0 - alpha * mean0;
  float ud = u.x * u.x + u.y * u.y + u.z * u.z + u.w * u.w;
  const float us = wave_sum(ud) - 2.f * u0 * u0;            // linner(u,u)
  const float un = sqrtf(fmaxf(us, CLAMP_F));
  const float t = acoshf(alpha) / un;
  float4 xT;
  xT.x = t * u.x;  xT.y = t * u.y;  xT.z = t * u.z;  xT.w = t * u.w;
  const float corr = (t * u0) / (1.f + mean0);              // x_T0/(1+mean0)
  xT.x -= corr * mv.x;  xT.y -= corr * mv.y;
  xT.z -= corr * mv.z;  xT.w -= corr * mv.w;
  if (lane == 0) xT.x -= corr;                              // - corr * e0
  return xT;
}

// =====================================================================
// Pass 2: Frechet variance partials (mean Euclidean norm of tangents).
// =====================================================================
__global__ void k_var_partial(const float* __restrict__ x, float* __restrict__ ws) {
  const int wave = (blockIdx.x * blockDim.x + threadIdx.x) >> 5;  // 0..BSZ*WPB_D-1
  const int lane = threadIdx.x & 31;
  const int b  = wave >> 9;            // / WPB_D (512)
  const int wb = wave & (WPB_D - 1);
  const float* __restrict__ meanb = ws + OFF_MEAN + (size_t)b * CH;
  const float4 mv   = *(const float4*)(meanb + 4 * lane);
  const float mean0 = meanb[0];
  const float* __restrict__ xb = x + (size_t)b * NPTS * CH;

  float vacc = 0.f;
  #pragma unroll
  for (int i = 0; i < PTS_W_D; ++i) {
    const float* __restrict__ xp = xb + (size_t)(wb * PTS_W_D + i) * CH;
    __builtin_prefetch(xp + CH + lane * 4, 0, 0);
    float4 xT = tangent_at_origin(xp, mv, mean0, lane);
    float nd = xT.x * xT.x + xT.y * xT.y + xT.z * xT.z + xT.w * xT.w;
    vacc += sqrtf(wave_sum(nd));       // Euclidean norm (jnp.linalg.norm)
  }
  if (lane == 0) ws[OFF_VPART + wave] = vacc;
}

// =====================================================================
// Finalize: g[b] = gamma / (var + eps). One block per batch.
// =====================================================================
__global__ void k_g_finalize(float* __restrict__ ws, const float* __restrict__ gamma) {
  __shared__ float sm[256];
  const int b = blockIdx.x, t = threadIdx.x;
  const float* __restrict__ vp = ws + OFF_VPART + (size_t)b * WPB_D;
  sm[t] = vp[t] + vp[t + 256];
  __syncthreads();
  #pragma unroll
  for (int off = 128; off > 0; off >>= 1) {
    if (t < off) sm[t] += sm[t + off];
    __syncthreads();
  }
  if (t == 0) ws[OFF_G + b] = gamma[0] / (sm[0] * (1.0f / (float)NPTS) + EPS_F);
}

// =====================================================================
// Pass 3: recompute tangent, scale, transport to beta, expmap, store.
// =====================================================================
__global__ void k_output(const float* __restrict__ x, const float* __restrict__ beta,
                         float* __restrict__ out, const float* __restrict__ ws) {
  const int wave = (blockIdx.x * blockDim.x + threadIdx.x) >> 5;
  const int lane = threadIdx.x & 31;
  const int b  = wave >> 9;
  const int wb = wave & (WPB_D - 1);
  const float* __restrict__ meanb = ws + OFF_MEAN + (size_t)b * CH;
  const float4 mv   = *(const float4*)(meanb + 4 * lane);
  const float mean0 = meanb[0];
  const float gs    = ws[OFF_G + b];
  const float4 bv   = *(const float4*)(beta + 4 * lane);
  const float beta0 = beta[0];
  const float* __restrict__ xb = x + (size_t)b * NPTS * CH;
  float* __restrict__ ob = out + (size_t)b * NPTS * CH;

  #pragma unroll
  for (int i = 0; i < PTS_W_D; ++i) {
    const int p = wb * PTS_W_D + i;
    const float* __restrict__ xp = xb + (size_t)p * CH;
    __builtin_prefetch(xp + CH + lane * 4, 0, 0);
    float4 xT = tangent_at_origin(xp, mv, mean0, lane);

    // scale by gamma/(var+eps)
    xT.x *= gs;  xT.y *= gs;  xT.z *= gs;  xT.w *= gs;

    // parallel transport origin -> beta: xT += (linner(beta,xT)/(1+beta0))*(beta+e0)
    const float xT0 = __shfl(xT.x, 0, 32);
    float bd = bv.x * xT.x + bv.y * xT.y + bv.z * xT.z + bv.w * xT.w;
    const float lb = wave_sum(bd) - 2.f * beta0 * xT0;
    const float coef = lb / (1.f + beta0);
    xT.x += coef * bv.x;  xT.y += coef * bv.y;
    xT.z += coef * bv.z;  xT.w += coef * bv.w;
    if (lane == 0) xT.x += coef;                 // + coef * e0

    // expmap at beta
    const float xT0f = __shfl(xT.x, 0, 32);
    float nd = xT.x * xT.x + xT.y * xT.y + xT.z * xT.z + xT.w * xT.w;
    const float nusq = wave_sum(nd) - 2.f * xT0f * xT0f;   // linner(xT,xT)
    const float nu = sqrtf(fmaxf(nusq, CLAMP_F));
    const float chv = coshf(nu);
    const float shv = sinhf(nu) / nu;
    float4 o;
    o.x = chv * bv.x + shv * xT.x;  o.y = chv * bv.y + shv * xT.y;
    o.z = chv * bv.z + shv * xT.z;  o.w = chv * bv.w + shv * xT.w;
    *(float4*)(ob + (size_t)p * CH + 4 * lane) = o;
  }
}

extern "C" void kernel_launch(void* const* d_in, const int* in_sizes, int n_in,
                              void* d_out, int out_size, void* d_ws, size_t ws_size,
                              hipStream_t stream) {
  (void)in_sizes; (void)n_in; (void)out_size; (void)ws_size;
  const float* x     = (const float*)d_in[0];
  const float* beta  = (const float*)d_in[1];
  const float* gamma = (const float*)d_in[2];
  float* out = (float*)d_out;
  float* ws  = (float*)d_ws;

  // Pass 1: mean partials (WMMA column sums): BSZ*WPB_B waves = 2048 -> 256 blocks
  k_mean_partial<<<dim3((BSZ * WPB_B * 32) / 256), dim3(256), 0, stream>>>(x, ws);
  // Finalize mean (project to hyperboloid)
  k_mean_finalize<<<dim3(BSZ), dim3(CH), 0, stream>>>(ws);
  // Pass 2: variance partials: BSZ*WPB_D waves = 32768 -> 4096 blocks
  k_var_partial<<<dim3((BSZ * WPB_D * 32) / 256), dim3(256), 0, stream>>>(x, ws);
  // Finalize gamma/(var+eps)
  k_g_finalize<<<dim3(BSZ), dim3(256), 0, stream>>>(ws, gamma);
  // Pass 3: output
  k_output<<<dim3((BSZ * WPB_D * 32) / 256), dim3(256), 0, stream>>>(x, beta, out, ws);
}